// ForwardDeformer_85993835201272
// MI455X (gfx1250) — compile-verified
//
#include <hip/hip_runtime.h>
#include <stdint.h>

typedef __attribute__((ext_vector_type(2))) float v2f;
typedef __attribute__((ext_vector_type(8))) float v8f;

#define JP     56      // J=55 padded to multiple of 4
#define KSTEPS 14      // 56 / 4
#define WAVES  8
#define TPB    256
#define BMAX   4

__global__ __launch_bounds__(TPB) void fwd_deformer_kernel(
    const float*   __restrict__ xc,          // [B,N,3]
    const float*   __restrict__ shape_off,   // [B,N,3]
    const float*   __restrict__ pose_off,    // [B,N,3]
    const float*   __restrict__ tfs,         // [B,J,4,4]
    const float*   __restrict__ tfs_inv,     // [1,J,4,4]
    const float*   __restrict__ poseoff_ori, // [1,N,3]
    const float*   __restrict__ lbsw,        // [1,N,J]
    const uint8_t* __restrict__ mask,        // [1,N] bool
    const float*   __restrict__ vox,         // [J,16,64,64]
    const float*   __restrict__ offk,        // [3]
    const float*   __restrict__ sclk,        // [3]
    float*         __restrict__ out_xd,      // [B,N,3]
    float*         __restrict__ out_w,       // [B,N,4,4]
    int N, int B, int J, int numTiles)
{
    // tfs staged transposed: sT[b][elem(0..15)][k(0..55)], zero padded beyond J
    __shared__ float sT[BMAX * 16 * JP];
    __shared__ float sTinv[16 * JP];
    __shared__ float sW[WAVES][1024];     // weights [16][56]; reused for C staging
    __shared__ float sPPw[WAVES][16 * 8]; // 8 trilinear corner weights / point
    __shared__ int   sPPo[WAVES][16 * 4]; // base, dx, dy, dz offsets / point

    const int tid = threadIdx.x;

    for (int i = tid; i < BMAX * 16 * JP; i += TPB) sT[i] = 0.f;
    for (int i = tid; i < 16 * JP; i += TPB) sTinv[i] = 0.f;
    __syncthreads();
    for (int i = tid; i < B * J * 16; i += TPB) {
        int b = i / (J * 16); int r = i - b * (J * 16);
        int j = r >> 4; int e = r & 15;
        sT[(b * 16 + e) * JP + j] = tfs[i];
    }
    for (int i = tid; i < J * 16; i += TPB) {
        int j = i >> 4; int e = i & 15;
        sTinv[e * JP + j] = tfs_inv[i];
    }
    __syncthreads();

    const int wave = tid >> 5;
    const int lane = tid & 31;
    const int tile = blockIdx.x * WAVES + wave;
    if (tile >= numTiles) return;          // wave-uniform: EXEC stays all-ones for WMMA
    const int b  = (tile * 16) / N;        // N % 16 == 0 -> tile never straddles batch
    const int n0 = tile * 16 - b * N;

    float* Wbuf = sW[wave];
    float* ppw  = sPPw[wave];
    int*   ppo  = sPPo[wave];

    // --- trilinear setup (grid_sample_3d, align_corners, border), lanes 0..15 ---
    if (lane < 16) {
        const int p = lane;
        const size_t gi = ((size_t)b * N + (n0 + p)) * 3;
        const float gx = (xc[gi + 0] + offk[0]) * sclk[0];
        const float gy = (xc[gi + 1] + offk[1]) * sclk[1];
        const float gz = (xc[gi + 2] + offk[2]) * sclk[2];
        const float ixf = fminf(fmaxf((gx + 1.f) * 31.5f, 0.f), 63.f); // W=64
        const float iyf = fminf(fmaxf((gy + 1.f) * 31.5f, 0.f), 63.f); // H=64
        const float izf = fminf(fmaxf((gz + 1.f) * 7.5f,  0.f), 15.f); // D=16
        const float fx0 = floorf(ixf), fy0 = floorf(iyf), fz0 = floorf(izf);
        const float wx = ixf - fx0, wy = iyf - fy0, wz = izf - fz0;
        const int x0 = (int)fx0, y0 = (int)fy0, z0 = (int)fz0;
        const int dxo = (x0 < 63) ? 1 : 0;
        const int dyo = (y0 < 63) ? 64 : 0;
        const int dzo = (z0 < 15) ? 4096 : 0;
        const float ux = 1.f - wx, uy = 1.f - wy, uz = 1.f - wz;
        ppw[p * 8 + 0] = uz * uy * ux;  ppw[p * 8 + 1] = uz * uy * wx;
        ppw[p * 8 + 2] = uz * wy * ux;  ppw[p * 8 + 3] = uz * wy * wx;
        ppw[p * 8 + 4] = wz * uy * ux;  ppw[p * 8 + 5] = wz * uy * wx;
        ppw[p * 8 + 6] = wz * wy * ux;  ppw[p * 8 + 7] = wz * wy * wx;
        ppo[p * 4 + 0] = z0 * 4096 + y0 * 64 + x0;
        ppo[p * 4 + 1] = dxo; ppo[p * 4 + 2] = dyo; ppo[p * 4 + 3] = dzo;
    }

    // --- cooperative weight-tile fill: W[16 points][56 joints] ---
    for (int it = 0; it < (16 * JP) / 32; ++it) {
        const int e = lane + it * 32;
        const int p = e / JP;
        const int j = e - p * JP;
        float w = 0.f;
        if (j < J) {
            const int n = n0 + p;
            if (mask[n]) {
                w = lbsw[(size_t)n * J + j];
            } else {
                const float* cp = vox + (size_t)j * 65536 + ppo[p * 4 + 0];
                const int dxo = ppo[p * 4 + 1];
                const int dyo = ppo[p * 4 + 2];
                const int dzo = ppo[p * 4 + 3];
                const float* pw = ppw + p * 8;
                w = cp[0]           * pw[0] + cp[dxo]             * pw[1]
                  + cp[dyo]         * pw[2] + cp[dyo + dxo]       * pw[3]
                  + cp[dzo]         * pw[4] + cp[dzo + dxo]       * pw[5]
                  + cp[dzo + dyo]   * pw[6] + cp[dzo + dyo + dxo] * pw[7];
            }
        }
        Wbuf[e] = w;
    }

    // --- WMMA: Cinv = W x Tinv^T-layout, Cfwd = W x T[b]; f32 16x16x4, 14 K-steps ---
    v8f cInv = {};
    v8f cFwd = {};
    const int halfk = (lane >> 4) * 2;        // A/B 32-bit layout: high lanes hold K+2,K+3
    const int r16   = lane & 15;
    const float* Arow  = Wbuf  + r16 * JP;               // A: row = point
    const float* BiRow = sTinv + r16 * JP;               // B: col = matrix element
    const float* BbRow = sT + (b * 16 + r16) * JP;
#pragma unroll
    for (int kk = 0; kk < KSTEPS; ++kk) {
        const int k0 = kk * 4 + halfk;
        v2f a  = *(const v2f*)(Arow  + k0);
        v2f bi = *(const v2f*)(BiRow + k0);
        v2f bb = *(const v2f*)(BbRow + k0);
        cInv = __builtin_amdgcn_wmma_f32_16x16x4_f32(false, a, false, bi,
                                                     (short)0, cInv, false, false);
        cFwd = __builtin_amdgcn_wmma_f32_16x16x4_f32(false, a, false, bb,
                                                     (short)0, cFwd, false, false);
    }

    // --- stage C tiles into LDS (row-major [point][16]) for per-lane epilogue ---
    {
        const int mbase = (lane >> 4) * 8;    // C layout: VGPR r -> M = r (+8 high lanes)
        const int ncol  = lane & 15;
#pragma unroll
        for (int r = 0; r < 8; ++r) {
            Wbuf[(mbase + r) * 16 + ncol]       = cInv[r];
            Wbuf[256 + (mbase + r) * 16 + ncol] = cFwd[r];
        }
    }

    // --- epilogue: one point per lane (lanes 0..15) ---
    if (lane < 16) {
        const int p = lane;
        const int n = n0 + p;
        float Mi[16], Mf[16];
#pragma unroll
        for (int i = 0; i < 16; ++i) {
            Mi[i] = Wbuf[p * 16 + i];
            Mf[i] = Wbuf[256 + p * 16 + i];
        }
        const size_t gi = ((size_t)b * N + n) * 3;
        const float xcx = xc[gi + 0], xcy = xc[gi + 1], xcz = xc[gi + 2];
        float xs[3];
#pragma unroll
        for (int i = 0; i < 3; ++i) {
            const float cano = Mi[i*4+0] * xcx + Mi[i*4+1] * xcy
                             + Mi[i*4+2] * xcz + Mi[i*4+3];
            xs[i] = cano - poseoff_ori[(size_t)n * 3 + i]
                  + shape_off[gi + i] + pose_off[gi + i];
        }
#pragma unroll
        for (int i = 0; i < 3; ++i) {
            out_xd[gi + i] = Mf[i*4+0] * xs[0] + Mf[i*4+1] * xs[1]
                           + Mf[i*4+2] * xs[2] + Mf[i*4+3];
        }
        float* ow = out_w + ((size_t)b * N + n) * 16;
#pragma unroll
        for (int i = 0; i < 4; ++i)
#pragma unroll
            for (int l = 0; l < 4; ++l)
                ow[i * 4 + l] = Mf[i*4+0] * Mi[0*4+l] + Mf[i*4+1] * Mi[1*4+l]
                              + Mf[i*4+2] * Mi[2*4+l] + Mf[i*4+3] * Mi[3*4+l];
    }
}

extern "C" void kernel_launch(void* const* d_in, const int* in_sizes, int n_in,
                              void* d_out, int out_size, void* d_ws, size_t ws_size,
                              hipStream_t stream) {
    (void)n_in; (void)out_size; (void)d_ws; (void)ws_size;
    const int J = in_sizes[4] / 16;              // tfs_inv = [1,J,4,4]
    const int B = in_sizes[3] / in_sizes[4];     // tfs = [B,J,4,4]
    const int N = in_sizes[5] / 3;               // poseoff_ori = [1,N,3]
    const int numTiles = (B * N + 15) / 16;
    const int blocks = (numTiles + WAVES - 1) / WAVES;

    float* out_xd = (float*)d_out;
    float* out_w  = out_xd + (size_t)B * N * 3;

    fwd_deformer_kernel<<<blocks, TPB, 0, stream>>>(
        (const float*)d_in[0], (const float*)d_in[1], (const float*)d_in[2],
        (const float*)d_in[3], (const float*)d_in[4], (const float*)d_in[5],
        (const float*)d_in[6], (const uint8_t*)d_in[7], (const float*)d_in[8],
        (const float*)d_in[9], (const float*)d_in[10],
        out_xd, out_w, N, B, J, numTiles);
}